// DiagWinAttention_62929860821588
// MI455X (gfx1250) — compile-verified
//
#include <hip/hip_runtime.h>
#include <math.h>

// ---------------------------------------------------------------------------
// DiagWinAttention fused kernel for gfx1250 (MI455X).
// One workgroup (256 threads = 8 wave32) per window.
// - All GEMMs: V_WMMA_F32_16X16X4_F32 from LDS-staged fp32 operands.
// - All global->LDS staging: GLOBAL_LOAD_ASYNC_TO_LDS_B128 (ASYNCcnt),
//   double-buffered so value/Wv/Wp loads overlap the k-GEMM / head loop.
// - k/v outputs: GLOBAL_STORE_ASYNC_FROM_LDS_B128 draining under compute.
// ---------------------------------------------------------------------------

namespace {
constexpr int kNW   = 4096;          // total windows
constexpr int kNWM  = 2048;          // mask windows
constexpr int kNP   = 64;            // patches per window
constexpr int kED   = 96;            // embed dim
constexpr int kNH   = 6;             // heads
constexpr int kCH   = 16;            // channels per head
constexpr float kScale = 0.25f;      // CH^-0.5
constexpr float kEps   = 1e-5f;
constexpr int kTok  = kNP * kED;                 // 6144 floats per window
constexpr int kTok16 = kTok * 4 / 16;            // 1536 16B chunks
constexpr int kW16   = kED * kED * 4 / 16;       // 2304 16B chunks
constexpr long long kOutStride = (long long)kNW * kTok;
}

typedef __attribute__((ext_vector_type(2))) float v2f;
typedef __attribute__((ext_vector_type(8))) float v8f;

// ---- async global<->LDS (ASYNCcnt) ----------------------------------------
__device__ __forceinline__ uint32_t lds_addr(const void* p) {
  // generic shared address: [63:32]=aperture, [31:0]=LDS offset
  return (uint32_t)(uintptr_t)p;
}
__device__ __forceinline__ void async_ld16(uint32_t lds, uint32_t voff,
                                           const void* sbase) {
  asm volatile("global_load_async_to_lds_b128 %0, %1, %2"
               :: "v"(lds), "v"(voff), "s"(sbase) : "memory");
}
__device__ __forceinline__ void async_st16(uint32_t voff, uint32_t lds,
                                           void* sbase) {
  asm volatile("global_store_async_from_lds_b128 %0, %1, %2"
               :: "v"(voff), "v"(lds), "s"(sbase) : "memory");
}
#define WAIT_ASYNC(imm) asm volatile("s_wait_asynccnt " #imm ::: "memory")

// chunked async copies (uniform instruction count across all waves)
__device__ __forceinline__ void acopy_in(void* ldsb, const void* g, int n16, int tid) {
  const uint32_t l = lds_addr(ldsb);
  for (int i = tid; i < n16; i += 256) async_ld16(l + i * 16, (uint32_t)(i * 16), g);
}
__device__ __forceinline__ void acopy_out(void* g, const void* ldsb, int n16, int tid) {
  const uint32_t l = lds_addr(ldsb);
  for (int i = tid; i < n16; i += 256) async_st16((uint32_t)(i * 16), l + i * 16, g);
}

// ---- WMMA helpers ----------------------------------------------------------
__device__ __forceinline__ v8f wmma4(v8f acc, v2f a, v2f b) {
  return __builtin_amdgcn_wmma_f32_16x16x4_f32(false, a, false, b, (short)0, acc,
                                               false, false);
}

// D(16x16) += A(16xK, row-major, stride lda) * B(Kx16, row-major, stride ldb)
__device__ __forceinline__ v8f gemm_nn(v8f acc, const float* __restrict__ A, int lda,
                                       const float* __restrict__ B, int ldb, int K,
                                       int lane) {
  const int r  = lane & 15;
  const int kb = (lane >> 4) << 1;
  for (int k0 = 0; k0 < K; k0 += 4) {
    v2f a, b;
    a[0] = A[r * lda + k0 + kb];
    a[1] = A[r * lda + k0 + kb + 1];
    b[0] = B[(k0 + kb) * ldb + r];
    b[1] = B[(k0 + kb + 1) * ldb + r];
    acc = wmma4(acc, a, b);
  }
  return acc;
}

// D(16x16) += A(16xK) * B(Kx16) with B transposed: B[k][n] = Bt[n*ldbt + k]
__device__ __forceinline__ v8f gemm_nt(v8f acc, const float* __restrict__ A, int lda,
                                       const float* __restrict__ Bt, int ldbt, int K,
                                       int lane) {
  const int r  = lane & 15;
  const int kb = (lane >> 4) << 1;
  for (int k0 = 0; k0 < K; k0 += 4) {
    v2f a, b;
    a[0] = A[r * lda + k0 + kb];
    a[1] = A[r * lda + k0 + kb + 1];
    b[0] = Bt[r * ldbt + k0 + kb];
    b[1] = Bt[r * ldbt + k0 + kb + 1];
    acc = wmma4(acc, a, b);
  }
  return acc;
}

__device__ __forceinline__ void store_c(float* __restrict__ D, int ldd, v8f c, int lane) {
  const int n  = lane & 15;
  const int mo = (lane >> 4) << 3;
#pragma unroll
  for (int i = 0; i < 8; ++i) D[(mo + i) * ldd + n] = c[i];
}

__device__ __forceinline__ float gelu_exact(float x) {
  return 0.5f * x * (1.0f + erff(x * 0.70710678118654752f));
}

__device__ __forceinline__ void ln_gelu_row96(float* row, const float* __restrict__ b,
                                              const float* __restrict__ g,
                                              const float* __restrict__ beta) {
  float mu = 0.f;
  for (int c = 0; c < kED; ++c) mu += row[c] + b[c];
  mu *= (1.0f / kED);
  float var = 0.f;
  for (int c = 0; c < kED; ++c) { float d = row[c] + b[c] - mu; var += d * d; }
  var *= (1.0f / kED);
  const float rs = rsqrtf(var + kEps);
  for (int c = 0; c < kED; ++c) {
    float y = (row[c] + b[c] - mu) * rs * g[c] + beta[c];
    row[c] = gelu_exact(y);
  }
}

__global__ __launch_bounds__(256, 1)
void diagwin_fused(const float* __restrict__ query, const float* __restrict__ keyp,
                   const float* __restrict__ valuep, const float* __restrict__ maskp,
                   const float* __restrict__ Wk, const float* __restrict__ bk,
                   const float* __restrict__ gk, const float* __restrict__ betak,
                   const float* __restrict__ Wv, const float* __restrict__ bv,
                   const float* __restrict__ gv, const float* __restrict__ betav,
                   const float* __restrict__ bias_table,
                   const float* __restrict__ Ws, const float* __restrict__ bs,
                   const float* __restrict__ gs, const float* __restrict__ betas,
                   const float* __restrict__ gn, const float* __restrict__ bn,
                   const float* __restrict__ Wp, const float* __restrict__ bp,
                   float* __restrict__ out) {
  __shared__ __align__(16) float s_q[kNP * kED];     // query (raw, then scaled)
  __shared__ __align__(16) float s_in[kNP * kED];    // key staging, then x96
  __shared__ __align__(16) float s_in2[kNP * kED];   // value staging
  __shared__ __align__(16) float s_k[kNP * kED];     // k MLP output
  __shared__ __align__(16) float s_v[kNP * kED];     // v MLP output
  __shared__ __align__(16) float s_w[kED * kED];     // Wk -> Wp staging
  __shared__ __align__(16) float s_w2[kED * kED];    // Wv staging
  __shared__ __align__(16) float s_attn[kNP * kNP];  // per-head logits / probs
  __shared__ __align__(16) float s_out[kNP * kCH];   // attn @ v per head
  __shared__ __align__(16) float s_sc[kNP * kCH];    // shortcut GEMM output
  __shared__ __align__(16) float s_ws[2 * kCH * kCH];// Ws (32x16)

  const int tid  = threadIdx.x;
  const int wave = tid >> 5;
  const int lane = tid & 31;
  const int w    = blockIdx.x;

  const float* qg = query  + (long long)w * kTok;
  const float* kg = keyp   + (long long)w * kTok;
  const float* vg = valuep + (long long)w * kTok;
  float* o1 = out + (long long)w * kTok;                   // qo
  float* o2 = out + kOutStride + (long long)w * kTok;      // rev(kh) == k
  float* o3 = out + 2 * kOutStride + (long long)w * kTok;  // rev(vh) == v

  // ---- stage 0: issue ALL async input loads ------------------------------
  // phase A (needed first): q(6) + Wk(9) + key(6) + Ws(1) = 22 issues/wave
  acopy_in(s_q, qg, kTok16, tid);
  acopy_in(s_w, Wk, kW16, tid);
  acopy_in(s_in, kg, kTok16, tid);
  { const int i = tid & 127;                               // 128 chunks, 1/thread
    async_ld16(lds_addr(s_ws) + i * 16, (uint32_t)(i * 16), Ws); }
  // phase B (needed after k-GEMM): value(6) + Wv(9) = 15 issues/wave
  acopy_in(s_in2, vg, kTok16, tid);
  acopy_in(s_w2, Wv, kW16, tid);

  WAIT_ASYNC(0xf);          // in-order completion => phase A resident
  __syncthreads();

  // scale q in place
  for (int i = tid; i < kTok / 4; i += 256) {
    float4 t = ((float4*)s_q)[i];
    t.x *= kScale; t.y *= kScale; t.z *= kScale; t.w *= kScale;
    ((float4*)s_q)[i] = t;
  }
  __syncthreads();

  // ---- k = GELU(LN(key @ Wk + bk)) ---------------------------------------
  for (int t = wave; t < 24; t += 8) {
    const int ti = t / 6, tj = t % 6;
    v8f acc = {};
    acc = gemm_nn(acc, s_in + ti * 16 * kED, kED, s_w + tj * 16, kED, kED, lane);
    store_c(s_k + ti * 16 * kED + tj * 16, kED, acc, lane);
  }
  __syncthreads();
  if (tid < kNP) ln_gelu_row96(s_k + tid * kED, bk, gk, betak);
  __syncthreads();

  WAIT_ASYNC(0x0);          // value + Wv resident
  __syncthreads();

  // overlap with v-GEMM / head loop: Wp load + k-output store, both async
  acopy_in(s_w, Wp, kW16, tid);       // s_w free (k-GEMM done)
  acopy_out(o2, s_k, kTok16, tid);    // s_k is read-only from here on

  // ---- v = GELU(LN(value @ Wv + bv)) -------------------------------------
  for (int t = wave; t < 24; t += 8) {
    const int ti = t / 6, tj = t % 6;
    v8f acc = {};
    acc = gemm_nn(acc, s_in2 + ti * 16 * kED, kED, s_w2 + tj * 16, kED, kED, lane);
    store_c(s_v + ti * 16 * kED + tj * 16, kED, acc, lane);
  }
  __syncthreads();
  if (tid < kNP) ln_gelu_row96(s_v + tid * kED, bv, gv, betav);
  __syncthreads();
  acopy_out(o3, s_v, kTok16, tid);    // drains under the head loop

  // ---- per-head attention -------------------------------------------------
  const float* mrow_base = maskp + (long long)(w % kNWM) * kNP * kNP;
  for (int h = 0; h < kNH; ++h) {
    // logits: q_h (64x16) @ k_h^T (16x64) -> 4x4 tiles of 16x16
    for (int t = wave; t < 16; t += 8) {
      const int ti = t >> 2, tj = t & 3;
      v8f acc = {};
      acc = gemm_nt(acc, s_q + ti * 16 * kED + h * kCH, kED,
                    s_k + tj * 16 * kED + h * kCH, kED, kCH, lane);
      store_c(s_attn + ti * 16 * kNP + tj * 16, kNP, acc, lane);
    }
    __syncthreads();

    // bias + mask + softmax, one thread per row
    if (tid < kNP) {
      const int m = tid, my = m >> 3, mx = m & 7;
      float* row = s_attn + m * kNP;
      const float* mrow = mrow_base + m * kNP;
      float mxv = -3.4e38f;
      for (int n = 0; n < kNP; ++n) {
        const int ny = n >> 3, nx = n & 7;
        const float bias = bias_table[((my - ny + 7) * 15 + (mx - nx + 7)) * kNH + h];
        const float val = row[n] + bias + mrow[n];
        row[n] = val;
        mxv = fmaxf(mxv, val);
      }
      float s = 0.f;
      for (int n = 0; n < kNP; ++n) { const float e = expf(row[n] - mxv); row[n] = e; s += e; }
      const float inv = 1.0f / s;
      for (int n = 0; n < kNP; ++n) row[n] *= inv;
    }
    __syncthreads();

    // out_h = attn @ v_h (64x16), then sc = concat(out_h, q_h) @ Ws (K=32)
    if (wave < 4) {
      const int ti = wave;
      v8f acc = {};
      acc = gemm_nn(acc, s_attn + ti * 16 * kNP, kNP, s_v + h * kCH, kED, kNP, lane);
      store_c(s_out + ti * 16 * kCH, kCH, acc, lane);
      v8f a2 = {};
      a2 = gemm_nn(a2, s_out + ti * 16 * kCH, kCH, s_ws, kCH, kCH, lane);
      a2 = gemm_nn(a2, s_q + ti * 16 * kED + h * kCH, kED, s_ws + kCH * kCH, kCH,
                   kCH, lane);
      store_c(s_sc + ti * 16 * kCH, kCH, a2, lane);
    }
    __syncthreads();

    // LN(16) + GELU, scatter into x96 at column block h*16
    if (tid < kNP) {
      const int p = tid;
      float vals[kCH];
      float mu = 0.f;
      for (int c = 0; c < kCH; ++c) { vals[c] = s_sc[p * kCH + c] + bs[c]; mu += vals[c]; }
      mu *= (1.0f / kCH);
      float var = 0.f;
      for (int c = 0; c < kCH; ++c) { const float d = vals[c] - mu; var += d * d; }
      var *= (1.0f / kCH);
      const float rs = rsqrtf(var + kEps);
      for (int c = 0; c < kCH; ++c) {
        const float y = (vals[c] - mu) * rs * gs[c] + betas[c];
        s_in[p * kED + h * kCH + c] = gelu_exact(y);
      }
    }
    __syncthreads();
  }

  // ---- qo = LN(x96) @ Wp + bp --------------------------------------------
  WAIT_ASYNC(0x0);          // Wp resident (o2/o3 stores also retired)
  __syncthreads();
  if (tid < kNP) {
    float* row = s_in + tid * kED;
    float mu = 0.f;
    for (int c = 0; c < kED; ++c) mu += row[c];
    mu *= (1.0f / kED);
    float var = 0.f;
    for (int c = 0; c < kED; ++c) { const float d = row[c] - mu; var += d * d; }
    var *= (1.0f / kED);
    const float rs = rsqrtf(var + kEps);
    for (int c = 0; c < kED; ++c) row[c] = (row[c] - mu) * rs * gn[c] + bn[c];
  }
  __syncthreads();

  for (int t = wave; t < 24; t += 8) {
    const int ti = t / 6, tj = t % 6;
    v8f acc = {};
    acc = gemm_nn(acc, s_in + ti * 16 * kED, kED, s_w + tj * 16, kED, kED, lane);
    const int n = lane & 15;
    const int mo = (lane >> 4) << 3;
    const float bb = bp[tj * 16 + n];
#pragma unroll
    for (int i = 0; i < 8; ++i)
      __builtin_nontemporal_store(acc[i] + bb,
                                  &o1[(ti * 16 + mo + i) * kED + tj * 16 + n]);
  }
}

extern "C" void kernel_launch(void* const* d_in, const int* in_sizes, int n_in,
                              void* d_out, int out_size, void* d_ws, size_t ws_size,
                              hipStream_t stream) {
  (void)in_sizes; (void)n_in; (void)out_size; (void)d_ws; (void)ws_size;
  const float* query      = (const float*)d_in[0];
  const float* keyp       = (const float*)d_in[1];
  const float* valuep     = (const float*)d_in[2];
  const float* maskp      = (const float*)d_in[3];
  const float* Wk         = (const float*)d_in[4];
  const float* bk         = (const float*)d_in[5];
  const float* gk         = (const float*)d_in[6];
  const float* betak      = (const float*)d_in[7];
  const float* Wv         = (const float*)d_in[8];
  const float* bv         = (const float*)d_in[9];
  const float* gv         = (const float*)d_in[10];
  const float* betav      = (const float*)d_in[11];
  const float* bias_table = (const float*)d_in[12];
  const float* Ws         = (const float*)d_in[13];
  const float* bs         = (const float*)d_in[14];
  const float* gs         = (const float*)d_in[15];
  const float* betas      = (const float*)d_in[16];
  const float* gn         = (const float*)d_in[17];
  const float* bn         = (const float*)d_in[18];
  const float* Wp         = (const float*)d_in[19];
  const float* bp         = (const float*)d_in[20];
  float* out = (float*)d_out;

  diagwin_fused<<<dim3(kNW), dim3(256), 0, stream>>>(
      query, keyp, valuep, maskp, Wk, bk, gk, betak, Wv, bv, gv, betav,
      bias_table, Ws, bs, gs, betas, gn, bn, Wp, bp, out);
}